// Int4WeightOnlyEmbedding_41523743817966
// MI455X (gfx1250) — compile-verified
//
#include <hip/hip_runtime.h>
#include <stdint.h>

// Int4 group-quantized embedding gather:
//   out[t, :] = (weight[x[t], :] - zp[x[t], g]) * scale[x[t], g],  g = channel/32
// V=128000, D=2048, G=32 (64 groups), B*S=32768 tokens.
// Memory-bound: ~540 MB moved -> ~23 us at 23.3 TB/s. Strategy: one block per
// token, async global->LDS b128 row gather (CDNA5 ASYNCcnt path), ds_load_b128
// consume, non-temporal b128 stores.

typedef int   v4i __attribute__((ext_vector_type(4)));
typedef float v4f __attribute__((ext_vector_type(4)));

#ifndef EMB_D
#define EMB_D 2048
#define EMB_G 32
#define EMB_NG (EMB_D / EMB_G)   // 64 groups per row
#endif

__global__ __launch_bounds__(256)
void Int4WeightOnlyEmbedding_41523743817966_kernel(
    const int*   __restrict__ weight,   // [V, D] int4 values in int32 containers
    const float* __restrict__ scale,    // [V, 64]
    const int*   __restrict__ zp,       // [V, 64]
    const int*   __restrict__ x,        // [B*S]
    float*       __restrict__ out,      // [B*S, D]
    int n_tok)
{
    __shared__ __align__(16) int smem[EMB_D];   // 8 KB: one dequant row staging buffer

    const int token = blockIdx.x;
    if (token >= n_tok) return;
    const int t = threadIdx.x;                  // 0..255

    // Uniform per block -> scalar load of the row index.
    const int idx = x[token];

    // ---- Async DMA the 8 KB weight row into LDS (CDNA5 async-to-LDS path) ----
    // GVS addressing: global = SADDR(row base) + VADDR(lane offset) + inst_offset
    //                 lds    = VDST(lane lds byte addr)            + inst_offset
    // Two b128 per lane: bytes [t*16, t*16+16) and the same +4096.
    const int* row = weight + (long long)idx * EMB_D;
    unsigned lds_addr = (unsigned)(uintptr_t)(&smem[0]) + (unsigned)(t * 16);
    int      goff     = t * 16;
    asm volatile(
        "global_load_async_to_lds_b128 %0, %1, %2\n\t"
        "global_load_async_to_lds_b128 %0, %1, %2 offset:4096"
        :
        : "v"(lds_addr), "v"(goff), "s"(row)
        : "memory");

    // ---- While the DMA is in flight, fetch the (tiny) group scale/zp ----
    // Chunk A = channels [t*4, t*4+4)        -> group t/8
    // Chunk B = channels [1024+t*4, ...+4)   -> group 32 + t/8
    const long long sbase = (long long)idx * EMB_NG;
    const int gA = t >> 3;
    const int gB = EMB_NG / 2 + (t >> 3);
    const float sA = scale[sbase + gA];
    const float sB = scale[sbase + gB];
    const float zA = (float)zp[sbase + gA];
    const float zB = (float)zp[sbase + gB];

    // Each lane consumes only bytes its own wave's async ops wrote -> waiting on
    // this wave's ASYNCcnt is sufficient (no workgroup barrier needed).
    asm volatile("s_wait_asynccnt 0" ::: "memory");

    // ---- Consume from LDS (ds_load_b128) and dequantize ----
    const v4i wA = ((const v4i*)smem)[t];
    const v4i wB = ((const v4i*)smem)[256 + t];

    v4f oA, oB;
    oA.x = ((float)wA.x - zA) * sA;
    oA.y = ((float)wA.y - zA) * sA;
    oA.z = ((float)wA.z - zA) * sA;
    oA.w = ((float)wA.w - zA) * sA;
    oB.x = ((float)wB.x - zB) * sB;
    oB.y = ((float)wB.y - zB) * sB;
    oB.z = ((float)wB.z - zB) * sB;
    oB.w = ((float)wB.w - zB) * sB;

    // ---- Non-temporal b128 stores: output (256 MB) is never re-read ----
    float* orow = out + (long long)token * EMB_D;
    __builtin_nontemporal_store(oA, (v4f*)orow + t);
    __builtin_nontemporal_store(oB, (v4f*)orow + 256 + t);
}

extern "C" void kernel_launch(void* const* d_in, const int* in_sizes, int n_in,
                              void* d_out, int out_size, void* d_ws, size_t ws_size,
                              hipStream_t stream) {
    const int*   weight = (const int*)  d_in[0];   // [V, D] int32 (int4 values)
    const float* scale  = (const float*)d_in[1];   // [V, D/G] fp32
    const int*   zp     = (const int*)  d_in[2];   // [V, D/G] int32
    const int*   x      = (const int*)  d_in[3];   // [B*S] int32
    float*       out    = (float*)d_out;           // [B*S, D] fp32

    const int n_tok = in_sizes[3];                 // B*S = 32768

    dim3 grid(n_tok);
    dim3 block(256);                               // 8 waves per token/block
    Int4WeightOnlyEmbedding_41523743817966_kernel<<<grid, block, 0, stream>>>(
        weight, scale, zp, x, out, n_tok);
}